// SparseAttention_89678917141071
// MI455X (gfx1250) — compile-verified
//
#include <hip/hip_runtime.h>

#define HC 128   // hidden channels
#define NH 8     // heads
// D = 16, HC = D*NH

typedef __attribute__((ext_vector_type(2))) float v2f;
typedef __attribute__((ext_vector_type(8))) float v8f;

// ---- monotone float <-> uint key for atomic segment-max ------------------
__device__ __forceinline__ unsigned float_key(float f) {
  unsigned u = __float_as_uint(f);
  return (u & 0x80000000u) ? ~u : (u | 0x80000000u);
}
__device__ __forceinline__ float key_float(unsigned k) {
  unsigned u = (k & 0x80000000u) ? (k & 0x7FFFFFFFu) : ~k;
  return __uint_as_float(u);
}

// ---- workspace init -------------------------------------------------------
__global__ void init_ws_kernel(unsigned* __restrict__ segkey,
                               float* __restrict__ denom,
                               float* __restrict__ agg, int n) {
  int gid = blockIdx.x * blockDim.x + threadIdx.x;
  if (gid < n * NH) { segkey[gid] = 0u; denom[gid] = 0.0f; }
  if (gid < n * HC) { agg[gid] = 0.0f; }
}

// ---- templated fp32 WMMA GEMM: C = A[M,K] @ B[K,N] ------------------------
// blockIdx.x: group of 8 row tiles (8 waves/block), blockIdx.y: 16-col tile.
// B column tile is staged once per block into LDS, pre-swizzled into the
// WMMA B-fragment layout (one v2f per lane per k-step) -> each fragment
// fetch is a single conflict-free ds_load_b64.
template <int K, int LDA, int LDB, int LDC>
__global__ __launch_bounds__(256) void wmma_gemm_f32(const float* __restrict__ A,
                                                     const float* __restrict__ B,
                                                     float* __restrict__ C,
                                                     int Mrows) {
  constexpr int KSTEPS = K / 4;
  __shared__ v2f Bs[KSTEPS * 32];                 // K=128 -> 8 KB

  const int tn = blockIdx.y;
#pragma unroll
  for (int idx = threadIdx.x; idx < KSTEPS * 32; idx += 256) {
    const int kk = idx >> 5, L = idx & 31;
    const int hf = L >> 4, lr2 = L & 15;
    const int ka = kk * 4 + 2 * hf;
    const int col = tn * 16 + lr2;
    v2f t;
    t.x = B[(size_t)ka * LDB + col];
    t.y = B[(size_t)(ka + 1) * LDB + col];
    Bs[idx] = t;
  }
  __syncthreads();

  const int lane = threadIdx.x & 31;
  const int tm = blockIdx.x * 8 + (threadIdx.x >> 5);
  if (tm * 16 >= Mrows) return;                   // wave-uniform; no barrier after

  const int lr = lane & 15;
  const int half = lane >> 4;
  int arow = tm * 16 + lr; if (arow >= Mrows) arow = Mrows - 1;
  const float* ap = A + (size_t)arow * LDA + 2 * half;  // 8B-aligned (LDA mult of 2, ka even)

  v8f acc = {};
#pragma unroll
  for (int kk = 0; kk < KSTEPS; ++kk) {
    const v2f a = *(const v2f*)(ap + kk * 4);
    acc = __builtin_amdgcn_wmma_f32_16x16x4_f32(false, a, false, Bs[kk * 32 + lane],
                                                (short)0, acc, false, false);
  }

  const int bcol = tn * 16 + lr;
#pragma unroll
  for (int v = 0; v < 8; ++v) {
    const int r = tm * 16 + v + 8 * half;
    if (r < Mrows) C[(size_t)r * LDC + bcol] = acc[v];
  }
}

// ---- fused edge kernel: bias GEMM (pr@Wb, never materialized) + logits ----
// One wave per 16 edges; 8 waves/block share the per-j Wb fragment tiles via
// LDS (all waves walk the same col tiles j=0..7). Indices are clamped (no
// EXEC-divergent branches) so EXEC is all-ones at every WMMA; barriers gate
// the LDS staging, so no early returns.
__global__ __launch_bounds__(256) void edge_bias_logit_kernel(
    const float* __restrict__ PR, const float* __restrict__ Wb,
    const float* __restrict__ qbuf, const float* __restrict__ kvbuf,
    const int* __restrict__ qidx, const int* __restrict__ kidx,
    float* __restrict__ logit_out, unsigned* __restrict__ segkey, int Medges) {
  constexpr int KSTEPS = HC / 4;                  // 32
  __shared__ v2f BmS[KSTEPS * 32];                // 8 KB
  __shared__ v2f BaS[KSTEPS * 32];                // 8 KB

  const int lane = threadIdx.x & 31;
  const int mtiles = (Medges + 15) >> 4;
  int tm = blockIdx.x * 8 + (threadIdx.x >> 5);
  if (tm >= mtiles) tm = mtiles - 1;              // duplicate tile: benign writes

  const int half = lane >> 4;
  const int lr = lane & 15;
  int arow = tm * 16 + lr; if (arow >= Medges) arow = Medges - 1;

  // A fragment (one paired_repr row, all K) hoisted to registers: 64 VGPRs
  v2f afrag[KSTEPS];
  const float* prow = PR + (size_t)arow * HC + 2 * half;
#pragma unroll
  for (int kk = 0; kk < KSTEPS; ++kk) afrag[kk] = *(const v2f*)(prow + kk * 4);

  float logacc[8];
#pragma unroll
  for (int v = 0; v < 8; ++v) logacc[v] = 0.0f;

  for (int j = 0; j < 8; ++j) {
    __syncthreads();                              // previous tile consumed
#pragma unroll
    for (int idx = threadIdx.x; idx < KSTEPS * 32; idx += 256) {
      const int kk = idx >> 5, L = idx & 31;
      const int hf = L >> 4, lr2 = L & 15;
      const int ka = kk * 4 + 2 * hf;
      const int col = j * 16 + lr2;               // 0..127
      v2f t1, t2;
      t1.x = Wb[(size_t)ka * 256 + col];
      t1.y = Wb[(size_t)(ka + 1) * 256 + col];
      t2.x = Wb[(size_t)ka * 256 + 128 + col];
      t2.y = Wb[(size_t)(ka + 1) * 256 + 128 + col];
      BmS[idx] = t1;
      BaS[idx] = t2;
    }
    __syncthreads();

    v8f bm = {};
    v8f ba = {};
#pragma unroll
    for (int kk = 0; kk < KSTEPS; ++kk) {
      bm = __builtin_amdgcn_wmma_f32_16x16x4_f32(false, afrag[kk], false,
                                                 BmS[kk * 32 + lane], (short)0,
                                                 bm, false, false);
      ba = __builtin_amdgcn_wmma_f32_16x16x4_f32(false, afrag[kk], false,
                                                 BaS[kk * 32 + lane], (short)0,
                                                 ba, false, false);
    }

    // consume tile j: per-edge gathers (q/kv buffers are L2-resident)
    const int col = j * 16 + lr;
#pragma unroll
    for (int v = 0; v < 8; ++v) {
      int e = tm * 16 + v + 8 * half;
      e = (e < Medges) ? e : (Medges - 1);        // clamp, no EXEC branch
      const int qi = qidx[e];
      const int ki = kidx[e];
      const float qv = qbuf[(size_t)qi * HC + col];
      const float kv = kvbuf[(size_t)ki * (2 * HC) + col];   // k = kv[:, :128]
      logacc[v] += qv * (kv * (1.0f + bm[v]) + ba[v]);
    }
  }

  const int h = lane & 7;
#pragma unroll
  for (int v = 0; v < 8; ++v) {
    logacc[v] += __shfl_xor(logacc[v], 8, 32);    // combine complementary d-halves
    const int e = tm * 16 + v + 8 * half;
    if (((lane & 8) == 0) && e < Medges) {
      logit_out[(size_t)e * NH + h] = logacc[v];
      atomicMax(&segkey[(size_t)qidx[e] * NH + h], float_key(logacc[v]));
    }
  }
}

// ---- segment exp-sum ------------------------------------------------------
__global__ void exp_denom_kernel(const float* __restrict__ logit,
                                 const int* __restrict__ qidx,
                                 const unsigned* __restrict__ segkey,
                                 float* __restrict__ denom, int Medges) {
  int gid = blockIdx.x * blockDim.x + threadIdx.x;
  if (gid >= Medges * NH) return;
  const int e = gid >> 3, h = gid & 7;
  const int qi = qidx[e];
  const float mx = key_float(segkey[qi * NH + h]);
  const float ex = __expf(logit[(size_t)e * NH + h] - mx);
  atomicAdd(&denom[qi * NH + h], ex);
}

// ---- scatter attn * v into agg -------------------------------------------
__global__ void scatter_agg_kernel(const float* __restrict__ logit,
                                   const int* __restrict__ qidx,
                                   const int* __restrict__ kidx,
                                   const unsigned* __restrict__ segkey,
                                   const float* __restrict__ denom,
                                   const float* __restrict__ kvbuf,
                                   float* __restrict__ agg, int Medges) {
  int gid = blockIdx.x * blockDim.x + threadIdx.x;     // M*128 = 51.2M < 2^31
  if (gid >= Medges * HC) return;
  const int e = gid >> 7, c = gid & 127, h = c & 7;
  const int qi = qidx[e];
  const float mx   = key_float(segkey[qi * NH + h]);
  const float attn = __expf(logit[(size_t)e * NH + h] - mx) / denom[qi * NH + h];
  const float ve   = kvbuf[(size_t)kidx[e] * (2 * HC) + HC + c];  // v = kv[:,128:]
  atomicAdd(&agg[(size_t)qi * HC + c], attn * ve);
}

// ---------------------------------------------------------------------------
extern "C" void kernel_launch(void* const* d_in, const int* in_sizes, int n_in,
                              void* d_out, int out_size, void* d_ws, size_t ws_size,
                              hipStream_t stream) {
  const float* query = (const float*)d_in[0];
  const float* key_  = (const float*)d_in[1];
  const int*   qidx  = (const int*)d_in[2];
  const int*   kidx  = (const int*)d_in[3];
  const float* pr    = (const float*)d_in[4];
  const float* Wq    = (const float*)d_in[5];
  const float* Wkv   = (const float*)d_in[6];
  const float* Wb    = (const float*)d_in[7];
  const float* Wo    = (const float*)d_in[8];

  const int N = in_sizes[0] / HC;
  const int M = in_sizes[2];

  float* out_result = (float*)d_out;                       // N*HC
  float* out_logit  = out_result + (size_t)N * HC;         // M*NH

  // workspace layout (floats): q | kv | segkey | denom | agg  (~106 MB)
  float*    q_buf  = (float*)d_ws;
  float*    kv_buf = q_buf + (size_t)N * HC;
  unsigned* segkey = (unsigned*)(kv_buf + (size_t)N * 2 * HC);
  float*    denom  = (float*)(segkey + (size_t)N * NH);
  float*    agg    = denom + (size_t)N * NH;

  const int TPB = 256;  // 8 waves/block (wave32)
  const int mtilesN = (N + 15) / 16;
  const int mtilesM = (M + 15) / 16;

  // 0) init accumulators
  init_ws_kernel<<<(N * HC + TPB - 1) / TPB, TPB, 0, stream>>>(segkey, denom, agg, N);

  // 1) q = query @ Wq   (N x 128)
  wmma_gemm_f32<HC, HC, HC, HC>
      <<<dim3((mtilesN + 7) / 8, HC / 16), TPB, 0, stream>>>(query, Wq, q_buf, N);

  // 2) kv = key @ Wkv   (N x 256)
  wmma_gemm_f32<HC, HC, 2 * HC, 2 * HC>
      <<<dim3((mtilesN + 7) / 8, 2 * HC / 16), TPB, 0, stream>>>(key_, Wkv, kv_buf, N);

  // 3) fused pr@Wb bias GEMM + per-head logits + segment max
  edge_bias_logit_kernel<<<(mtilesM + 7) / 8, TPB, 0, stream>>>(
      pr, Wb, q_buf, kv_buf, qidx, kidx, out_logit, segkey, M);

  // 4) segment exp-sum
  exp_denom_kernel<<<(M * NH + TPB - 1) / TPB, TPB, 0, stream>>>(
      out_logit, qidx, segkey, denom, M);

  // 5) scatter attn * v
  scatter_agg_kernel<<<(M * HC + TPB - 1) / TPB, TPB, 0, stream>>>(
      out_logit, qidx, kidx, segkey, denom, kv_buf, agg, M);

  // 6) result = agg @ Wo
  wmma_gemm_f32<HC, HC, HC, HC>
      <<<dim3((mtilesN + 7) / 8, HC / 16), TPB, 0, stream>>>(agg, Wo, out_result, N);
}